// GAT_41652592837290
// MI455X (gfx1250) — compile-verified
//
#include <hip/hip_runtime.h>
#include <hip/hip_bf16.h>
#include <math.h>

#define N_NODES 100000
#define N_EDGES 1600000
#define F_IN    256
#define HEADS   4
#define DIM     32
#define F_OUT   (HEADS * DIM)   // 128
#define NEG_SLOPE 0.2f

// Interleaved LDS layout for W: per k-quad q, two halves of (256 data + 32 pad)
// floats. Element W[k][n] lives at q*576 + ((k>>1)&1)*288 + n*2 + (k&1).
// A lane's (W[kk][n], W[kk+1][n]) pair is contiguous -> single ds_load_b64,
// and the +288 half offset puts lanes 16-31 on banks 32..63 (conflict-free).
#define LDS_QUAD_STRIDE 576
#define LDS_HALF_STRIDE 288
#define LDS_W_FLOATS    (64 * LDS_QUAD_STRIDE)   // 36864 floats = 144 KB

typedef float v2f __attribute__((ext_vector_type(2)));
typedef float v8f __attribute__((ext_vector_type(8)));

// ---------- monotonic float <-> uint map for atomic max on floats ----------
__device__ __forceinline__ unsigned ford(float f) {
    unsigned u = __float_as_uint(f);
    return (u & 0x80000000u) ? ~u : (u | 0x80000000u);
}
__device__ __forceinline__ float unford(unsigned u) {
    u = (u & 0x80000000u) ? (u & 0x7fffffffu) : ~u;
    return __uint_as_float(u);
}

// =====================================================================
// Kernel 1: feat = h @ W  via V_WMMA_F32_16X16X4_F32 (full fp32 precision)
//   block = 256 threads = 8 wave32; each wave computes a 16x128 strip.
//   W staged in 144 KB dynamic LDS in a k-pair-interleaved, bank-padded
//   layout so every B-tile fetch is one conflict-free ds_load_b64.
// =====================================================================
__global__ void gat_gemm_wmma(const float* __restrict__ h,
                              const float* __restrict__ W,
                              float* __restrict__ feat, int n) {
    extern __shared__ float sW[];            // LDS_W_FLOATS floats
    const int tid = threadIdx.x;

    // cooperative load of W into interleaved LDS layout
    for (int i = tid; i < F_IN * F_OUT; i += 256) {
        const int k = i >> 7;                // i / 128
        const int nn = i & 127;
        const int idx = (k >> 2) * LDS_QUAD_STRIDE +
                        ((k >> 1) & 1) * LDS_HALF_STRIDE +
                        (nn << 1) + (k & 1);
        sW[idx] = W[i];
    }
    __syncthreads();

    const int wave = tid >> 5;
    const int lane = tid & 31;
    const int row0 = blockIdx.x * 128 + wave * 16;

    // A-matrix 16x4 f32 layout: lanes 0-15 -> M=lane, K = k0+{0,1};
    //                           lanes 16-31 -> M=lane-16, K = k0+{2,3}
    const int m    = lane & 15;
    const int half = lane >> 4;              // 0 or 1 (selects k-pair)
    long rowA = row0 + m;
    if (rowA > n - 1) rowA = n - 1;          // clamp: keep EXEC all-ones for WMMA
    const float* hrow = h + rowA * (long)F_IN;

    // per-lane base into LDS for B fetches (tile offset added as immediate)
    const float* bBase = sW + half * LDS_HALF_STRIDE + ((lane & 15) << 1);

    v8f acc[8] = {};                         // 8 N-tiles of 16 cols each

    for (int k0 = 0; k0 < F_IN; k0 += 4) {
        v2f a = *(const v2f*)(hrow + k0 + (half << 1));
        __builtin_prefetch(hrow + k0 + 64, 0, 3);   // global_prefetch_b8 ahead on h
        const float* bq = bBase + (k0 >> 2) * LDS_QUAD_STRIDE;
        #pragma unroll
        for (int t = 0; t < 8; ++t) {
            // tile t -> columns t*16.. ; lane's pair sits 32 floats apart per tile
            v2f b = *(const v2f*)(bq + (t << 5));
            acc[t] = __builtin_amdgcn_wmma_f32_16x16x4_f32(
                false, a, false, b, (short)0, acc[t], false, false);
        }
    }

    // C/D layout: VGPR j holds M=j (lanes 0-15) / M=j+8 (lanes 16-31), N=lane&15
    const int mBase = row0 + (half << 3);
    #pragma unroll
    for (int t = 0; t < 8; ++t) {
        const int nn = (t << 4) + (lane & 15);
        #pragma unroll
        for (int j = 0; j < 8; ++j) {
            const int row = mBase + j;
            if (row < n) feat[(long)row * F_OUT + nn] = acc[t][j];
        }
    }
}

// =====================================================================
// Kernel 2: e_src[n,h] = <feat[n,h,:], a_src[h,:]>, same for e_dst.
//   One wave per node; lane d, shfl_xor tree reduction (wave32).
// =====================================================================
__global__ void gat_attn_halves(const float* __restrict__ feat,
                                const float* __restrict__ a_src,
                                const float* __restrict__ a_dst,
                                float* __restrict__ e_src,
                                float* __restrict__ e_dst, int n) {
    const int node = (blockIdx.x * blockDim.x + threadIdx.x) >> 5;
    const int lane = threadIdx.x & 31;
    if (node >= n) return;
    const float* f = feat + (long)node * F_OUT;
    #pragma unroll
    for (int hh = 0; hh < HEADS; ++hh) {
        const float v = f[hh * DIM + lane];
        float s = v * a_src[hh * DIM + lane];
        float d = v * a_dst[hh * DIM + lane];
        #pragma unroll
        for (int off = 16; off > 0; off >>= 1) {
            s += __shfl_xor(s, off, 32);
            d += __shfl_xor(d, off, 32);
        }
        if (lane == 0) {
            e_src[node * HEADS + hh] = s;
            e_dst[node * HEADS + hh] = d;
        }
    }
}

// =====================================================================
// Kernel 3: per-edge logits (leaky relu) + segmented max by dst (atomic).
// =====================================================================
__global__ void gat_logits_segmax(const float* __restrict__ e_src,
                                  const float* __restrict__ e_dst,
                                  const int* __restrict__ src_idx,
                                  const int* __restrict__ dst_idx,
                                  float* __restrict__ logits,
                                  unsigned* __restrict__ segmax) {
    const long t = (long)blockIdx.x * blockDim.x + threadIdx.x;
    if (t >= (long)N_EDGES * HEADS) return;
    const int e  = (int)(t >> 2);
    const int hh = (int)(t & 3);
    const int s  = src_idx[e];
    const int d  = dst_idx[e];
    float l = e_src[s * HEADS + hh] + e_dst[d * HEADS + hh];
    l = (l > 0.0f) ? l : NEG_SLOPE * l;
    logits[t] = l;
    atomicMax(&segmax[d * HEADS + hh], ford(l));
}

// =====================================================================
// Kernel 4: ex = exp(logit - segmax[dst]); segmented sum by dst (atomic).
//   ex overwrites the logits buffer.
// =====================================================================
__global__ void gat_exp_segsum(const int* __restrict__ dst_idx,
                               const unsigned* __restrict__ segmax,
                               float* __restrict__ logits_ex,
                               float* __restrict__ segsum) {
    const long t = (long)blockIdx.x * blockDim.x + threadIdx.x;
    if (t >= (long)N_EDGES * HEADS) return;
    const int e  = (int)(t >> 2);
    const int hh = (int)(t & 3);
    const int d  = dst_idx[e];
    const float mx = unford(segmax[d * HEADS + hh]);
    const float ex = expf(logits_ex[t] - mx);
    logits_ex[t] = ex;
    atomicAdd(&segsum[d * HEADS + hh], ex);
}

// =====================================================================
// Kernel 5: scatter aggregate: out[dst, f] += feat[src, f] * alpha(e, f/32)
//   thread per (edge, feature); fire-and-forget global_atomic_add_f32.
// =====================================================================
__global__ void gat_scatter(const float* __restrict__ feat,
                            const int* __restrict__ src_idx,
                            const int* __restrict__ dst_idx,
                            const float* __restrict__ ex,
                            const float* __restrict__ segsum,
                            float* __restrict__ out) {
    const long t = (long)blockIdx.x * blockDim.x + threadIdx.x;
    if (t >= (long)N_EDGES * F_OUT) return;
    const int e  = (int)(t >> 7);
    const int f  = (int)(t & 127);
    const int hh = f >> 5;
    const int s  = src_idx[e];
    const int d  = dst_idx[e];
    const float alpha = ex[e * HEADS + hh] / segsum[d * HEADS + hh];
    atomicAdd(&out[(long)d * F_OUT + f], feat[(long)s * F_OUT + f] * alpha);
}

// =====================================================================
// Kernel 6: out = elu(out + bias), in place.
// =====================================================================
__global__ void gat_bias_elu(float* __restrict__ out,
                             const float* __restrict__ bias, int n) {
    const long t = (long)blockIdx.x * blockDim.x + threadIdx.x;
    if (t >= (long)n * F_OUT) return;
    const int f = (int)(t & 127);
    float x = out[t] + bias[f];
    out[t] = (x > 0.0f) ? x : (expf(x) - 1.0f);
}

// =====================================================================
extern "C" void kernel_launch(void* const* d_in, const int* in_sizes, int n_in,
                              void* d_out, int out_size, void* d_ws, size_t ws_size,
                              hipStream_t stream) {
    const float* h     = (const float*)d_in[0];   // [N, 256]
    const float* W     = (const float*)d_in[1];   // [256, 128]
    const float* a_src = (const float*)d_in[2];   // [4, 32]
    const float* a_dst = (const float*)d_in[3];   // [4, 32]
    const float* bias  = (const float*)d_in[4];   // [4, 32]
    const int*   sidx  = (const int*)d_in[5];     // [E]
    const int*   didx  = (const int*)d_in[6];     // [E]
    float* out = (float*)d_out;                   // [N, 128]

    // workspace layout (bytes), all 16B-aligned
    char* ws = (char*)d_ws;
    float*    feat   = (float*)(ws);                                     // 51,200,000 B
    float*    e_src  = (float*)(ws + 51200000);                          //  1,600,000 B
    float*    e_dst  = (float*)(ws + 52800000);                          //  1,600,000 B
    unsigned* segmax = (unsigned*)(ws + 54400000);                       //  1,600,000 B
    float*    segsum = (float*)(ws + 56000000);                          //  1,600,000 B
    float*    ex     = (float*)(ws + 57600000);                          // 25,600,000 B

    // zero accumulators (graph-capture-safe async memsets)
    hipMemsetAsync(segmax, 0, (size_t)N_NODES * HEADS * sizeof(unsigned), stream);
    hipMemsetAsync(segsum, 0, (size_t)N_NODES * HEADS * sizeof(float), stream);
    hipMemsetAsync(out,    0, (size_t)N_NODES * F_OUT * sizeof(float), stream);

    // 1) GEMM via fp32 WMMA, W staged in 144 KB interleaved dynamic LDS
    {
        dim3 grid((N_NODES + 127) / 128);
        gat_gemm_wmma<<<grid, 256, LDS_W_FLOATS * sizeof(float), stream>>>(h, W, feat, N_NODES);
    }
    // 2) attention halves (wave per node)
    {
        long waves = N_NODES;
        dim3 grid((unsigned)((waves * 32 + 255) / 256));
        gat_attn_halves<<<grid, 256, 0, stream>>>(feat, a_src, a_dst, e_src, e_dst, N_NODES);
    }
    // 3) logits + seg max
    {
        long total = (long)N_EDGES * HEADS;
        dim3 grid((unsigned)((total + 255) / 256));
        gat_logits_segmax<<<grid, 256, 0, stream>>>(e_src, e_dst, sidx, didx, ex, segmax);
    }
    // 4) exp + seg sum
    {
        long total = (long)N_EDGES * HEADS;
        dim3 grid((unsigned)((total + 255) / 256));
        gat_exp_segsum<<<grid, 256, 0, stream>>>(didx, segmax, ex, segsum);
    }
    // 5) weighted scatter into out
    {
        long total = (long)N_EDGES * F_OUT;
        dim3 grid((unsigned)((total + 255) / 256));
        gat_scatter<<<grid, 256, 0, stream>>>(feat, sidx, didx, ex, segsum, out);
    }
    // 6) bias + ELU epilogue
    {
        long total = (long)N_NODES * F_OUT;
        dim3 grid((unsigned)((total + 255) / 256));
        gat_bias_elu<<<grid, 256, 0, stream>>>(out, bias, N_NODES);
    }
}